// GraphLevelPooling_22256520528423
// MI455X (gfx1250) — compile-verified
//
#include <hip/hip_runtime.h>
#include <math.h>

typedef float v2f __attribute__((ext_vector_type(2)));
typedef float v8f __attribute__((ext_vector_type(8)));

#define H 128
#define NUM_GRAPHS 512

// ---------------------------------------------------------------------------
// Phase 1: scatter edge attributes onto nodes. One float4 per thread
// (coalesced 16B loads of the 1.64 GB edge stream), 4 float atomics into the
// 25.6 MB node accumulator which stays resident in the 192 MB L2.
// ---------------------------------------------------------------------------
__global__ __launch_bounds__(256) void scatter_edges_kernel(
    const float* __restrict__ attr, const int* __restrict__ idx,
    float* __restrict__ node_acc, long long total_f4) {
  long long i = (long long)blockIdx.x * blockDim.x + threadIdx.x;
  if (i >= total_f4) return;
  long long e = i >> 5;          // edge id (32 float4 per edge row of 128)
  int c4 = (int)(i & 31);        // which float4 within the row
  const float4 v = ((const float4*)attr)[i];
  long long node = idx[e];
  float* dst = node_acc + node * H + c4 * 4;
  atomicAdd(dst + 0, v.x);
  atomicAdd(dst + 1, v.y);
  atomicAdd(dst + 2, v.z);
  atomicAdd(dst + 3, v.w);
}

// ---------------------------------------------------------------------------
// Per-graph node counts (for the scatter-mean).
// ---------------------------------------------------------------------------
__global__ __launch_bounds__(256) void count_kernel(const int* __restrict__ batch,
                                                    float* __restrict__ cnt, int N) {
  int n = blockIdx.x * blockDim.x + threadIdx.x;
  if (n < N) atomicAdd(&cnt[batch[n]], 1.0f);
}

// ---------------------------------------------------------------------------
// Phase 2: node_emb = ELU((node_acc + edge_attr0) @ W^T + b), pooled into
// per-graph sums via atomics. Each wave32 computes a 16-node x 128-col tile
// using V_WMMA_F32_16X16X4_F32 (fp32 kept end-to-end; GEMM is only 1.6 GFLOP
// so precision beats throughput here). W (64 KB) is staged in LDS per block.
//
// Fragment layouts per CDNA5 ISA 7.12.2:
//   A 16x4 f32 : lane m=lane&15 -> row M=m; lane half kh=lane>>4 picks
//                K = {k+2kh, k+2kh+1} in VGPR0/1.
//   B 4x16 f32 : VGPR0 lane-half rows K=k+2kh, VGPR1 rows K=k+2kh+1,
//                N = lane&15.  B[kk][n] = W[n + 16t][kk]  (D = X * W^T).
//   D 16x16 f32: VGPR r, lane -> (M = r + 8*kh, N = m).
// ---------------------------------------------------------------------------
__global__ __launch_bounds__(256) void gemm_elu_pool_kernel(
    const float* __restrict__ node_acc, const float* __restrict__ ea0,
    const float* __restrict__ Wm, const float* __restrict__ bias,
    const int* __restrict__ batch, int N, float* __restrict__ graph_sum) {
  __shared__ float Wlds[H * H];   // 64 KB of the 320 KB WGP LDS

  const int tid = threadIdx.x;
  for (int i = tid; i < (H * H) / 4; i += 256)
    ((float4*)Wlds)[i] = ((const float4*)Wm)[i];
  __syncthreads();

  const int wave = tid >> 5;
  const int lane = tid & 31;
  const int m    = lane & 15;
  const int kh   = lane >> 4;
  const int row0 = (blockIdx.x * 8 + wave) * 16;

  const v8f zero = {0.f, 0.f, 0.f, 0.f, 0.f, 0.f, 0.f, 0.f};
  v8f c[8];
#pragma unroll
  for (int t = 0; t < 8; ++t) c[t] = zero;

  for (int k = 0; k < H; k += 4) {
    v2f a;
    const int row = row0 + m;
    if (row < N) {
      const long long base = (long long)row * H + k + 2 * kh;
      a.x = node_acc[base] + ea0[base];
      a.y = node_acc[base + 1] + ea0[base + 1];
    } else {
      a.x = 0.0f;
      a.y = 0.0f;
    }
#pragma unroll
    for (int t = 0; t < 8; ++t) {
      const int j = t * 16 + m;           // output column this lane supplies
      v2f bb;
      bb.x = Wlds[j * H + k + 2 * kh];
      bb.y = Wlds[j * H + k + 2 * kh + 1];
      c[t] = __builtin_amdgcn_wmma_f32_16x16x4_f32(
          /*neg_a=*/false, a, /*neg_b=*/false, bb,
          /*c_mod=*/(short)0, c[t], /*reuse_a=*/false, /*reuse_b=*/false);
    }
  }

  // Epilogue: bias + ELU(alpha=1) + atomic scatter-sum into graph buckets.
#pragma unroll
  for (int t = 0; t < 8; ++t) {
    const int col = t * 16 + m;
    const float bv = bias[col];
#pragma unroll
    for (int r = 0; r < 8; ++r) {
      const int node = row0 + r + 8 * kh;
      if (node < N) {
        float v = c[t][r] + bv;
        v = (v > 0.0f) ? v : (expf(v) - 1.0f);
        atomicAdd(&graph_sum[(long long)batch[node] * H + col], v);
      }
    }
  }
}

// ---------------------------------------------------------------------------
// Phase 3: scatter-mean finalize.
// ---------------------------------------------------------------------------
__global__ __launch_bounds__(256) void finalize_kernel(
    const float* __restrict__ graph_sum, const float* __restrict__ cnt,
    float* __restrict__ out) {
  int i = blockIdx.x * blockDim.x + threadIdx.x;
  if (i < NUM_GRAPHS * H) {
    float c = cnt[i >> 7];
    out[i] = graph_sum[i] / fmaxf(c, 1.0f);
  }
}

extern "C" void kernel_launch(void* const* d_in, const int* in_sizes, int n_in,
                              void* d_out, int out_size, void* d_ws, size_t ws_size,
                              hipStream_t stream) {
  const float* ea0   = (const float*)d_in[0];  // (N, 128)
  const float* ea1   = (const float*)d_in[1];  // (E, 128)
  const float* ea2   = (const float*)d_in[2];  // (E, 128)
  const int*   idx1  = (const int*)d_in[3];    // (E,)
  const int*   idx2  = (const int*)d_in[4];    // (E,)
  const int*   batch = (const int*)d_in[5];    // (N,) sorted
  // d_in[6] = num_nodes scalar (unused; derived from in_sizes)
  const float* Wm    = (const float*)d_in[7];  // (128, 128)
  const float* bias  = (const float*)d_in[8];  // (128,)

  const int N = in_sizes[0] / H;
  const int E = in_sizes[1] / H;

  // Workspace layout: [node_acc N*128 | graph_sum 512*128 | cnt 512]
  float* node_acc  = (float*)d_ws;
  float* graph_sum = node_acc + (size_t)N * H;
  float* cnt       = graph_sum + (size_t)NUM_GRAPHS * H;

  const size_t zero_floats = (size_t)N * H + (size_t)NUM_GRAPHS * H + NUM_GRAPHS;
  hipMemsetAsync(node_acc, 0, zero_floats * sizeof(float), stream);

  // Phase 1: edge scatter (dominant: ~1.64 GB streamed, L2-resident atomics)
  {
    long long total_f4 = (long long)E * 32;
    int thr = 256;
    long long blocks = (total_f4 + thr - 1) / thr;
    scatter_edges_kernel<<<(int)blocks, thr, 0, stream>>>(ea1, idx1, node_acc, total_f4);
    scatter_edges_kernel<<<(int)blocks, thr, 0, stream>>>(ea2, idx2, node_acc, total_f4);
  }

  // Per-graph counts
  count_kernel<<<(N + 255) / 256, 256, 0, stream>>>(batch, cnt, N);

  // Phase 2: WMMA fp32 GEMM + ELU + pooling (128 nodes per 256-thread block)
  {
    int blocks = (N + 127) / 128;
    gemm_elu_pool_kernel<<<blocks, 256, 0, stream>>>(node_acc, ea0, Wm, bias,
                                                     batch, N, graph_sum);
  }

  // Phase 3: mean
  finalize_kernel<<<(NUM_GRAPHS * H + 255) / 256, 256, 0, stream>>>(graph_sum, cnt,
                                                                    (float*)d_out);
}